// SKL_42674795053910
// MI455X (gfx1250) — compile-verified
//
#include <hip/hip_runtime.h>
#include <hip/hip_bf16.h>

typedef __attribute__((ext_vector_type(2))) float v2f;
typedef __attribute__((ext_vector_type(8))) float v8f;

#define BSZ   8
#define QN    30
#define DN    1500
#define EN    768
#define NKER  11
#define WN    736      // (1500-30)/2 + 1
#define WINSZ 30
#define QPAD  32
#define DPAD  1504
#define DTILES 94      // 1504/16

// ---------------------------------------------------------------------------
// Kernel 1: inverse L2 norms for query rows (B*Q) and document rows (B*D).
// One block per row, 256 threads.
// ---------------------------------------------------------------------------
__global__ void skl_rownorm(const float* __restrict__ qe, const float* __restrict__ de,
                            float* __restrict__ qinv, float* __restrict__ dinv) {
    int row = blockIdx.x;
    const float* src;
    float* dst;
    if (row < BSZ * QN) { src = qe + (size_t)row * EN; dst = qinv + row; }
    else { int r = row - BSZ * QN; src = de + (size_t)r * EN; dst = dinv + r; }

    int t = threadIdx.x;
    float ss = 0.0f;
    for (int i = t; i < EN; i += 256) { float v = src[i]; ss += v * v; }
    __shared__ float red[8];
    #pragma unroll
    for (int off = 16; off > 0; off >>= 1) ss += __shfl_down(ss, off, 32);
    int lane = t & 31, wid = t >> 5;
    if (lane == 0) red[wid] = ss;
    __syncthreads();
    if (t == 0) {
        float s = 0.0f;
        #pragma unroll
        for (int i = 0; i < 8; ++i) s += red[i];
        *dst = 1.0f / fmaxf(sqrtf(s), 1e-13f);
    }
}

// ---------------------------------------------------------------------------
// Kernel 2: cos[b, q, d] = (qn . dn) via V_WMMA_F32_16X16X4_F32.
// One wave computes one 16x16 (q x d) tile; K looped in steps of 4.
// Padded rows handled by scale=0 so EXEC stays all-ones (WMMA requirement).
// A layout (16x4 f32): lanes 0-15 rows M, VGPR0/1 = K {0,1}; lanes 16-31 K {2,3}.
// B layout (4x16 f32): mirrored; row K striped across lanes within a VGPR.
// C/D layout: VGPR v: lanes 0-15 -> M=v, lanes 16-31 -> M=v+8, N=lane%16.
// ---------------------------------------------------------------------------
__global__ void skl_cos_wmma(const float* __restrict__ qe, const float* __restrict__ de,
                             const float* __restrict__ qinv, const float* __restrict__ dinv,
                             float* __restrict__ cosm) {
    const int wid  = threadIdx.x >> 5;
    const int lane = threadIdx.x & 31;
    const int dt = blockIdx.x * 8 + wid;   // d-tile
    const int qt = blockIdx.y;             // q-tile (0..1)
    const int b  = blockIdx.z;
    if (dt >= DTILES) return;              // wave-uniform exit

    const int half = lane >> 4;
    const int mn   = lane & 15;
    const int qrow = qt * 16 + mn;
    const int drow = dt * 16 + mn;
    const int qr = (qrow < QN) ? qrow : 0;
    const int dr = (drow < DN) ? drow : 0;
    const float ascale = (qrow < QN) ? qinv[b * QN + qrow] : 0.0f;
    const float bscale = (drow < DN) ? dinv[b * DN + drow] : 0.0f;
    const float* ap = qe + ((size_t)b * QN + qr) * EN + 2 * half;
    const float* bp = de + ((size_t)b * DN + dr) * EN + 2 * half;

    v8f acc = {};
    for (int k = 0; k < EN; k += 4) {
        v2f a, bb;
        a.x  = ap[k]     * ascale;
        a.y  = ap[k + 1] * ascale;
        bb.x = bp[k]     * bscale;
        bb.y = bp[k + 1] * bscale;
        acc = __builtin_amdgcn_wmma_f32_16x16x4_f32(
            /*neg_a=*/false, a, /*neg_b=*/false, bb,
            /*c_mod=*/(short)0, acc, /*reuse_a=*/false, /*reuse_b=*/false);
    }

    float* out = cosm + ((size_t)b * QPAD + qt * 16) * DPAD + dt * 16;
    const int mbase = half * 8;
    #pragma unroll
    for (int v = 0; v < 8; ++v)
        out[(size_t)(mbase + v) * DPAD + mn] = acc[v];
}

// ---------------------------------------------------------------------------
// Kernel 3: per (b, w): window-pool 11 RBF kernels over 30 doc positions for
// all 30 queries, saturation transform, sum over q, dot with dense_w.
// Block = 352 threads (11 waves); thread t<330 -> (q = t/11, k = t%11).
// ---------------------------------------------------------------------------
__global__ void skl_score(const float* __restrict__ cosm, const float* __restrict__ dmask,
                          const float* __restrict__ qmask, const float* __restrict__ qidf,
                          const float* __restrict__ mu, const float* __restrict__ sigma,
                          const float* __restrict__ sw1, const float* __restrict__ sb1,
                          const float* __restrict__ sw2, const float* __restrict__ sb2,
                          const float* __restrict__ sw3, const float* __restrict__ sb3,
                          const float* __restrict__ dw, float* __restrict__ scores) {
    const int w = blockIdx.x;
    const int b = blockIdx.y;
    const int t = threadIdx.x;
    const bool active = t < QN * NKER;
    const int q = t / NKER;
    const int k = t % NKER;

    float val = 0.0f;
    if (active) {
        const float muk = mu[k];
        const float sg  = sigma[k];
        const float is2 = 1.0f / (2.0f * sg * sg);
        const float* crow = cosm + ((size_t)b * QPAD + q) * DPAD + 2 * w;
        const float* dm   = dmask + b * DN + 2 * w;
        float pk = 0.0f;
        int len = 0;
        #pragma unroll 5
        for (int j = 0; j < WINSZ; ++j) {
            float m = dm[j];
            float c = crow[j];
            float diff = c - muk;
            pk += m * __expf(-diff * diff * is2);
            len += (m != 0.0f) ? 1 : 0;
        }
        float idf = fmaxf(qidf[b * QN + q], 0.0f);
        float lf = (float)len;
        float s1 = sw1[0] * idf + sw1[1] * lf + sb1[0];
        float s2 = 1.0f / (sw2[0] * idf + sw2[1] * lf + sb2[0]);
        float s3 = sw3[0] * idf + sw3[1] * lf + sb3[0];
        float x = fmaxf(pk, 1e-10f);
        float sp = s1 * __powf(x, s2) - s3;
        sp *= qmask[b * QN + q] * ((len > 0) ? 1.0f : 0.0f);
        val = sp * dw[k];
    }
    // reduce sum over all 352 threads
    __shared__ float red[11];
    float v = val;
    #pragma unroll
    for (int off = 16; off > 0; off >>= 1) v += __shfl_down(v, off, 32);
    int lane = t & 31, wid = t >> 5;
    if (lane == 0) red[wid] = v;
    __syncthreads();
    if (t == 0) {
        float s = 0.0f;
        #pragma unroll
        for (int i = 0; i < 11; ++i) s += red[i];
        scores[b * WN + w] = s;
    }
}

// ---------------------------------------------------------------------------
// Kernel 4: per batch: zero->-9900 substitution, TOP_K=3 argmax with +-15
// suppression, neighbor gather (idx, +-1, +-2), dot with chunk_scoring.
// ---------------------------------------------------------------------------
__global__ void skl_topk(const float* __restrict__ scores, const float* __restrict__ cs,
                         float* __restrict__ out) {
    const int b = blockIdx.x;
    const int t = threadIdx.x;
    __shared__ float s0[WN];
    __shared__ float mprs[WN];
    __shared__ float rv[256];
    __shared__ int   ri[256];
    __shared__ int   bestArr[3];

    for (int i = t; i < WN; i += 256) {
        float sc = scores[b * WN + i];
        float s = (sc == 0.0f) ? -9900.0f : sc;
        s0[i] = s;
        mprs[i] = s;
    }
    __syncthreads();

    for (int c = 0; c < 3; ++c) {
        float bv = -3.0e38f;
        int bi = WN;
        for (int i = t; i < WN; i += 256) {
            float v = mprs[i];
            if (v > bv || (v == bv && i < bi)) { bv = v; bi = i; }
        }
        rv[t] = bv; ri[t] = bi;
        __syncthreads();
        for (int s = 128; s > 0; s >>= 1) {
            if (t < s) {
                if (rv[t + s] > rv[t] || (rv[t + s] == rv[t] && ri[t + s] < ri[t])) {
                    rv[t] = rv[t + s]; ri[t] = ri[t + s];
                }
            }
            __syncthreads();
        }
        int best = ri[0];
        if (t == 0) bestArr[c] = best;
        for (int i = t; i < WN; i += 256) {
            int d = i - best; if (d < 0) d = -d;
            if (d < WINSZ / 2) mprs[i] = -10001.0f - (float)c;   // |r-best| < 15
        }
        __syncthreads();
    }

    if (t == 0) {
        const int offs[5] = {0, -1, 1, -2, 2};
        float acc = 0.0f;
        #pragma unroll
        for (int o = 0; o < 5; ++o)
            #pragma unroll
            for (int c = 0; c < 3; ++c) {
                int idx = bestArr[c] + offs[o];
                idx = idx < 0 ? 0 : (idx > WN - 1 ? WN - 1 : idx);
                float v = s0[idx];
                if (v <= -9900.0f) v = 0.0f;
                acc += v * cs[o * 3 + c];
            }
        out[b] = acc;
    }
}

// ---------------------------------------------------------------------------
extern "C" void kernel_launch(void* const* d_in, const int* in_sizes, int n_in,
                              void* d_out, int out_size, void* d_ws, size_t ws_size,
                              hipStream_t stream) {
    const float* qe    = (const float*)d_in[0];   // (8,30,768)
    const float* de    = (const float*)d_in[1];   // (8,1500,768)
    const float* qmask = (const float*)d_in[2];   // (8,30)
    const float* dmask = (const float*)d_in[3];   // (8,1500)
    const float* qidf  = (const float*)d_in[4];   // (8,30,1)
    // d_in[5] document_idfs: unused by reference
    const float* mu    = (const float*)d_in[6];   // (11)
    const float* sigma = (const float*)d_in[7];   // (11)
    const float* sw1   = (const float*)d_in[8];
    const float* sb1   = (const float*)d_in[9];
    const float* sw2   = (const float*)d_in[10];
    const float* sb2   = (const float*)d_in[11];
    const float* sw3   = (const float*)d_in[12];
    const float* sb3   = (const float*)d_in[13];
    const float* dw    = (const float*)d_in[14];  // (1,11)
    const float* cs    = (const float*)d_in[15];  // (1,15)
    float* out = (float*)d_out;                   // (8)

    float* ws = (float*)d_ws;
    float* qinv   = ws;                        // 240 floats (padded to 256)
    float* dinv   = ws + 256;                  // 12000 floats
    float* cosm   = ws + 12288;                // 8*32*1504 = 385024 floats
    float* scores = ws + 12288 + (size_t)BSZ * QPAD * DPAD;  // 8*736 floats

    skl_rownorm<<<dim3(BSZ * (QN + DN)), 256, 0, stream>>>(qe, de, qinv, dinv);
    skl_cos_wmma<<<dim3((DTILES + 7) / 8, 2, BSZ), 256, 0, stream>>>(qe, de, qinv, dinv, cosm);
    skl_score<<<dim3(WN, BSZ), 352, 0, stream>>>(cosm, dmask, qmask, qidf, mu, sigma,
                                                 sw1, sb1, sw2, sb2, sw3, sb3, dw, scores);
    skl_topk<<<dim3(BSZ), 256, 0, stream>>>(scores, cs, out);
}